// Soft_DTW_73607149519243
// MI455X (gfx1250) — compile-verified
//
#include <hip/hip_runtime.h>
#include <hip/hip_bf16.h>
#include <stdint.h>

// Problem constants (from the reference: B=8, T1=256, T2=1024, C=128)
#define T1C 256
#define T2C 1024
#define CC  128
#define DD  (T1C + T2C - 1)   // 1279
#define BIGF 10000000.0f
#define WPF  1.0f
#define TEMPF 0.01f

typedef int v4i __attribute__((ext_vector_type(4)));
typedef __attribute__((address_space(1))) v4i* gv4p;   // global (AS1) v4i*
typedef __attribute__((address_space(3))) v4i* lv4p;   // LDS    (AS3) v4i*

// ---------------------------------------------------------------------------
// CDNA5 async global->LDS helpers (ASYNCcnt-tracked data movement path)
// ---------------------------------------------------------------------------
__device__ __forceinline__ void async_copy_b128(void* lds_dst, const void* gsrc) {
#if defined(__has_builtin) && __has_builtin(__builtin_amdgcn_global_load_async_to_lds_b128)
  __builtin_amdgcn_global_load_async_to_lds_b128(
      (gv4p)(unsigned long long)gsrc,   // AS1 v4i*
      (lv4p)lds_dst,                    // AS3 v4i*
      0, 0);
#else
  unsigned loff = (unsigned)(unsigned long long)
      (__attribute__((address_space(3))) char*)lds_dst;
  unsigned long long ga = (unsigned long long)gsrc;
  asm volatile("global_load_async_to_lds_b128 %0, %1, off"
               :: "v"(loff), "v"(ga) : "memory");
#endif
}

__device__ __forceinline__ void wait_async0() {
#if defined(__has_builtin) && __has_builtin(__builtin_amdgcn_s_wait_asynccnt)
  __builtin_amdgcn_s_wait_asynccnt(0);
#else
  asm volatile("s_wait_asynccnt 0" ::: "memory");
#endif
}

// ---------------------------------------------------------------------------
// Kernel 1: pairwise L1 cost matrix  cost[b, t2, t1] = mean_c |A - B|
// 32x32 tile per 256-thread block; tiles staged to LDS with async b128 loads.
// ---------------------------------------------------------------------------
__global__ __launch_bounds__(256)
void Soft_DTW_cost_kernel(const float* __restrict__ A,   // (B,T1,C)
                          const float* __restrict__ Bm,  // (B,T2,C)
                          float* __restrict__ cost)      // (B,T2,T1)
{
  const int b    = blockIdx.z;
  const int t1_0 = blockIdx.x * 32;
  const int t2_0 = blockIdx.y * 32;

  __shared__ float sA[32][CC];   // 16 KB
  __shared__ float sB[32][CC];   // 16 KB

  const float4* gA4 = (const float4*)(A  + ((size_t)b * T1C + t1_0) * CC);
  const float4* gB4 = (const float4*)(Bm + ((size_t)b * T2C + t2_0) * CC);
  float4* sA4 = (float4*)&sA[0][0];
  float4* sB4 = (float4*)&sB[0][0];

  const int tid  = threadIdx.x;
  const int base = tid * 4;               // 1024 float4 per tile / 256 thr
#pragma unroll
  for (int i = 0; i < 4; ++i) {
    async_copy_b128(&sA4[base + i], &gA4[base + i]);
    async_copy_b128(&sB4[base + i], &gB4[base + i]);
  }
  wait_async0();
  __syncthreads();

  const int tx = tid & 15;                // t1 sub-tile (2 cols)
  const int ty = tid >> 4;                // t2 sub-tile (2 rows)
  float acc00 = 0.f, acc01 = 0.f, acc10 = 0.f, acc11 = 0.f;

#pragma unroll 4
  for (int c = 0; c < CC; c += 4) {
    float4 a0 = *(const float4*)&sA[tx * 2 + 0][c];
    float4 a1 = *(const float4*)&sA[tx * 2 + 1][c];
    float4 b0 = *(const float4*)&sB[ty * 2 + 0][c];
    float4 b1 = *(const float4*)&sB[ty * 2 + 1][c];
    acc00 += fabsf(a0.x-b0.x)+fabsf(a0.y-b0.y)+fabsf(a0.z-b0.z)+fabsf(a0.w-b0.w);
    acc01 += fabsf(a1.x-b0.x)+fabsf(a1.y-b0.y)+fabsf(a1.z-b0.z)+fabsf(a1.w-b0.w);
    acc10 += fabsf(a0.x-b1.x)+fabsf(a0.y-b1.y)+fabsf(a0.z-b1.z)+fabsf(a0.w-b1.w);
    acc11 += fabsf(a1.x-b1.x)+fabsf(a1.y-b1.y)+fabsf(a1.z-b1.z)+fabsf(a1.w-b1.w);
  }

  const float inv = 1.0f / (float)CC;
  const int t1a = t1_0 + tx * 2, t2a = t2_0 + ty * 2;
  float* outr0 = cost + ((size_t)b * T2C + t2a + 0) * T1C + t1a;
  float* outr1 = cost + ((size_t)b * T2C + t2a + 1) * T1C + t1a;
  outr0[0] = acc00 * inv; outr0[1] = acc01 * inv;
  outr1[0] = acc10 * inv; outr1[1] = acc11 * inv;
}

// ---------------------------------------------------------------------------
// Kernel 2: soft-DTW DP. One wave32 per batch; 8 cells per lane; the only
// cross-cell dependency crosses lanes via __shfl_up -> zero barriers.
// Cost rows are software-pipelined 2 diagonals ahead (hides L2 latency).
// Loop runs only to d == di (rows past di never feed the readout).
// ---------------------------------------------------------------------------
__global__ __launch_bounds__(32)
void Soft_DTW_dp_kernel(const float* __restrict__ cost,  // (B,T2,T1)
                        const int* __restrict__ len_a,
                        const int* __restrict__ len_b,
                        float* __restrict__ partial)     // (B)
{
  const int b    = blockIdx.x;
  const int lane = threadIdx.x;
  const float* cb = cost + (size_t)b * T2C * T1C;

  const float K  = 1.4426950408889634f / TEMPF;  // log2(e)/temp
  const float NT = 0.6931471805599453f * TEMPF;  // temp*ln(2)

  const int la = len_a[b];
  const int lb = len_b[b];
  const int di = la + lb - 2;            // final diagonal (readout row)
  const int lane_out = (la - 1) >> 3;
  const int k_out    = (la - 1) & 7;

  const int j0 = lane * 8;               // first cell index (0-based) this lane

  float pc[8], pcp[8];
#pragma unroll
  for (int k = 0; k < 8; ++k) { pc[k] = BIGF; pcp[k] = BIGF; }

  // preload diag 0 and 1 cost rows (clamped skew gather)
  float c0[8], c1[8], c2[8];
#pragma unroll
  for (int k = 0; k < 8; ++k) {
    int i0 = min(max(0 - (j0 + k), 0), T2C - 1);
    int i1 = min(max(1 - (j0 + k), 0), T2C - 1);
    c0[k] = cb[(size_t)i0 * T1C + j0 + k];
    c1[k] = cb[(size_t)i1 * T1C + j0 + k];
  }

  for (int d = 0; d <= di; ++d) {
    // prefetch diag d+2
    int dn = d + 2 < DD ? d + 2 : DD - 1;
#pragma unroll
    for (int k = 0; k < 8; ++k) {
      int idx = min(max(dn - (j0 + k), 0), T2C - 1);
      c2[k] = cb[(size_t)idx * T1C + j0 + k];
    }

    // neighbor (j-1) values from previous lane's slot 7
    float pcL  = __shfl_up(pc[7],  1);
    float pcpL = __shfl_up(pcp[7], 1);
    if (lane == 0) { pcL = BIGF; pcpL = (d == 0) ? 0.0f : BIGF; }

    float nv[8];
#pragma unroll
    for (int k = 0; k < 8; ++k) {
      float v0 = (k == 0) ? pcpL : pcp[k - 1];           // pcp[j-1]
      float v1 = pc[k] + WPF;                            // pc[j] + wp
      float v2 = ((k == 0) ? pcL : pc[k - 1]) + WPF;     // pc[j-1] + wp
      float m  = fminf(v0, fminf(v1, v2));
      float s  = exp2f((m - v0) * K) + exp2f((m - v1) * K) + exp2f((m - v2) * K);
      nv[k] = c0[k] + m - NT * __log2f(s);
    }
#pragma unroll
    for (int k = 0; k < 8; ++k) { pcp[k] = pc[k]; pc[k] = nv[k]; }
#pragma unroll
    for (int k = 0; k < 8; ++k) { c0[k] = c1[k]; c1[k] = c2[k]; }
  }

  // After the loop, pc[] holds diagonal di; single post-loop extract.
  if (lane == lane_out) {
    float r = pc[0];
#pragma unroll
    for (int k = 1; k < 8; ++k) r = (k == k_out) ? pc[k] : r;
    partial[b] = r;
  }
}

// ---------------------------------------------------------------------------
// Kernel 3: deterministic fixed-order sum of per-batch losses
// ---------------------------------------------------------------------------
__global__ void Soft_DTW_sum_kernel(const float* __restrict__ partial, int n,
                                    float* __restrict__ out)
{
  if (threadIdx.x == 0 && blockIdx.x == 0) {
    float s = 0.0f;
    for (int i = 0; i < n; ++i) s += partial[i];
    out[0] = s;
  }
}

// ---------------------------------------------------------------------------
extern "C" void kernel_launch(void* const* d_in, const int* in_sizes, int n_in,
                              void* d_out, int out_size, void* d_ws, size_t ws_size,
                              hipStream_t stream) {
  const float* fea_a = (const float*)d_in[0];   // (B,T1,C) f32
  const int*   len_a = (const int*)  d_in[1];   // (B)      i32
  const float* fea_b = (const float*)d_in[2];   // (B,T2,C) f32
  const int*   len_b = (const int*)  d_in[3];   // (B)      i32

  const int B = in_sizes[1];                    // 8

  float* cost    = (float*)d_ws;                                  // B*T2*T1 f32 (8 MB)
  float* partial = (float*)((char*)d_ws + (size_t)B * T2C * T1C * sizeof(float));

  dim3 cgrid(T1C / 32, T2C / 32, B);
  Soft_DTW_cost_kernel<<<cgrid, 256, 0, stream>>>(fea_a, fea_b, cost);
  Soft_DTW_dp_kernel<<<B, 32, 0, stream>>>(cost, len_a, len_b, partial);
  Soft_DTW_sum_kernel<<<1, 32, 0, stream>>>(partial, B, (float*)d_out);
}